// HistoryAwareAnchorRouter_7705171329192
// MI455X (gfx1250) — compile-verified
//
#include <hip/hip_runtime.h>
#include <math.h>

// HistoryAwareAnchorRouter — CDNA5 (gfx1250) implementation.
// Heavy math (u-projection GEMM and the T x T gram for compactness) runs on
// V_WMMA_F32_16X16X4_F32 (exact f32 accumulate). k and v GEMMs are folded
// away algebraically (scores = u.(Wk^T q), z = (sum_t y_t u_t) Wv^T).
// k_compact stages B-tiles of u into LDS with GLOBAL_LOAD_ASYNC_TO_LDS_B128
// (double-buffered, s_wait_asynccnt-throttled) so the 4 waves of a block
// share one copy instead of 4x redundant global reads.

#define B_   16
#define T_   2048
#define INDIM 256
#define DU   128
#define DA   64
#define DM   128
#define LST  2
#define KSEL 16.0f
#define EPSV 1e-6f
#define GRU_IN 133   // DU + 4 + 1
#define LDSROW 132   // padded row stride (floats): 32 distinct LDS banks

typedef float v2f __attribute__((ext_vector_type(2)));
typedef float v4f __attribute__((ext_vector_type(4)));
typedef float v8f __attribute__((ext_vector_type(8)));
typedef int   v4i __attribute__((vector_size(4 * sizeof(int))));

#define AS1 __attribute__((address_space(1)))
#define AS3 __attribute__((address_space(3)))

#if __has_builtin(__builtin_amdgcn_global_load_async_to_lds_b128)
#define HAVE_ASYNC_LDS 1
#else
#define HAVE_ASYNC_LDS 0
#endif

#if HAVE_ASYNC_LDS
#if __has_builtin(__builtin_amdgcn_s_wait_asynccnt)
#define WAIT_ASYNC(n) __builtin_amdgcn_s_wait_asynccnt(n)
#else
#define WAIT_ASYNC(n) asm volatile("s_wait_asynccnt %0" ::"i"(n) : "memory")
#endif
#endif

// ---------------- init: zero prev_sel and memory ----------------
__global__ void k_init(float* __restrict__ prev, float* __restrict__ mem) {
  int i = blockIdx.x * blockDim.x + threadIdx.x;
  if (i < B_ * T_) prev[i] = 0.f;
  if (i < B_ * DM) mem[i] = 0.f;
}

// ---------------- u = x @ W_u^T + b_u  (WMMA f32 16x16x4) ----------------
// Grid: (B*T/16)*(DU/16) waves, 4 waves/block. One wave -> one 16x16 tile.
__global__ void __launch_bounds__(128) k_proj_u(
    const float* __restrict__ x, const float* __restrict__ Wu,
    const float* __restrict__ bu, float* __restrict__ u) {
  int wave = (blockIdx.x * blockDim.x + threadIdx.x) >> 5;
  int lane = threadIdx.x & 31;
  int mt = wave >> 3;          // 0 .. B*T/16 - 1
  int nt = wave & 7;           // 0 .. 7   (DU/16)
  int n16 = lane & 15;
  int khalf = lane >> 4;       // 0/1 -> K pairs {0,1} / {2,3}

  const float* arow = x + (size_t)(mt * 16 + n16) * INDIM + khalf * 2;
  const float* brow = Wu + (size_t)(nt * 16 + n16) * INDIM + khalf * 2;

  v8f acc = {};
  for (int kk = 0; kk < INDIM; kk += 4) {
    v2f a = *(const v2f*)(arow + kk);
    v2f b = *(const v2f*)(brow + kk);
    acc = __builtin_amdgcn_wmma_f32_16x16x4_f32(false, a, false, b,
                                                (short)0, acc, false, false);
  }
  int rbase = mt * 16 + 8 * khalf;
  int col = nt * 16 + n16;
  float bias = bu[col];
#pragma unroll
  for (int v = 0; v < 8; ++v)
    u[(size_t)(rbase + v) * DU + col] = acc[v] + bias;
}

// ---------------- sq[b,t] = ||u_bt||^2 ----------------
__global__ void k_sq(const float* __restrict__ u, float* __restrict__ sq) {
  int i = blockIdx.x * blockDim.x + threadIdx.x;
  if (i >= B_ * T_) return;
  const float* p = u + (size_t)i * DU;
  float s = 0.f;
  for (int d = 0; d < DU; d += 4) {
    v4f w = *(const v4f*)(p + d);
    s += w.x * w.x + w.y * w.y + w.z * w.z + w.w * w.w;
  }
  sq[i] = s;
}

// ---------------- r[b,:] = W_k^T (W_q memory[b]) ----------------
__global__ void __launch_bounds__(128) k_qr(
    const float* __restrict__ mem, const float* __restrict__ Wq,
    const float* __restrict__ Wk, float* __restrict__ r) {
  __shared__ float ms[DM];
  __shared__ float qs[DA];
  int b = blockIdx.x, tid = threadIdx.x;
  ms[tid] = mem[b * DM + tid];
  __syncthreads();
  if (tid < DA) {
    float s = 0.f;
    for (int d = 0; d < DM; ++d) s += ms[d] * Wq[tid * DM + d];
    qs[tid] = s;
  }
  __syncthreads();
  float s = 0.f;
  for (int a = 0; a < DA; ++a) s += qs[a] * Wk[a * DU + tid];
  r[b * DU + tid] = s;
}

// ---------------- scores -> soft selection -> stats row-reductions ----------
__global__ void __launch_bounds__(1024) k_select(
    const float* __restrict__ u, const float* __restrict__ r,
    const float* __restrict__ b_pos, const float* __restrict__ log_temp,
    float* __restrict__ prev, float* __restrict__ ycur,
    float* __restrict__ stats, float* __restrict__ out, int stage) {
  __shared__ float ysh[T_];
  __shared__ float red[1024];
  __shared__ float rsh[DU];
  int b = blockIdx.x, tid = threadIdx.x;
  if (tid < DU) rsh[tid] = r[b * DU + tid];
  __syncthreads();

  float temp = expf(log_temp[0]);
  temp = fminf(fmaxf(temp, 0.1f), 10.f);
  const float inv_scale = 0.125f;  // 1/sqrt(DA)

  for (int t = tid; t < T_; t += 1024) {
    const float* up = u + ((size_t)b * T_ + t) * DU;
    float s = 0.f;
    for (int d = 0; d < DU; d += 4) {
      v4f w = *(const v4f*)(up + d);
      s += w.x * rsh[d] + w.y * rsh[d + 1] + w.z * rsh[d + 2] + w.w * rsh[d + 3];
    }
    s = s * inv_scale + b_pos[t] - prev[b * T_ + t];   // COV_GAMMA = 1
    ysh[t] = 1.f / (1.f + expf(-s / temp));
  }
  __syncthreads();

  // budget (pre-damping sum)
  red[tid] = ysh[tid] + ysh[tid + 1024];
  __syncthreads();
  for (int s = 512; s > 0; s >>= 1) { if (tid < s) red[tid] += red[tid + s]; __syncthreads(); }
  float budget = fmaxf(red[0], 1e-6f);
  float fct = fminf(KSEL / budget, 1.f);
  __syncthreads();
  ysh[tid] *= fct; ysh[tid + 1024] *= fct;
  __syncthreads();

  // refractory damping, sequential d = 1, 2 (wrap-around roll)
  for (int d = 1; d <= 2; ++d) {
    float a0 = ysh[tid],        s0 = ysh[(tid + d) & (T_ - 1)];
    float a1 = ysh[tid + 1024], s1 = ysh[(tid + 1024 + d) & (T_ - 1)];
    float n0 = a0 * fminf(2.f / (1.f + a0 + s0), 1.f);
    float n1 = a1 * fminf(2.f / (1.f + a1 + s1), 1.f);
    __syncthreads();
    ysh[tid] = n0; ysh[tid + 1024] = n1;
    __syncthreads();
  }
  if (tid == 0) ysh[0] = 0.f;
  __syncthreads();

  float y0 = ysh[tid], y1 = ysh[tid + 1024];

  // total
  red[tid] = y0 + y1;
  __syncthreads();
  for (int s = 512; s > 0; s >>= 1) { if (tid < s) red[tid] += red[tid + s]; __syncthreads(); }
  float rawsum = red[0];
  float total = fmaxf(rawsum, EPSV);
  __syncthreads();

  // sum y * pos
  red[tid] = y0 * (float)tid + y1 * (float)(tid + 1024);
  __syncthreads();
  for (int s = 512; s > 0; s >>= 1) { if (tid < s) red[tid] += red[tid + s]; __syncthreads(); }
  float meanp = red[0] / total;
  __syncthreads();

  // entropy
  float yn0 = y0 / total, yn1 = y1 / total;
  red[tid] = yn0 * logf(fmaxf(yn0, EPSV)) + yn1 * logf(fmaxf(yn1, EPSV));
  __syncthreads();
  for (int s = 512; s > 0; s >>= 1) { if (tid < s) red[tid] += red[tid + s]; __syncthreads(); }
  float entsum = red[0];
  __syncthreads();

  // variance of position
  float d0 = (float)tid - meanp, d1 = (float)(tid + 1024) - meanp;
  red[tid] = y0 * d0 * d0 + y1 * d1 * d1;
  __syncthreads();
  for (int s = 512; s > 0; s >>= 1) { if (tid < s) red[tid] += red[tid + s]; __syncthreads(); }
  float varsum = red[0];

  if (tid == 0) {
    stats[b * 8 + 0] = total;
    stats[b * 8 + 1] = rawsum / (float)T_;                 // coverage
    stats[b * 8 + 2] = -entsum;                            // entropy
    stats[b * 8 + 3] = sqrtf(fmaxf(varsum / total, EPSV)); // spacing
    stats[b * 8 + 4] = 0.f;                                // compact accumulator
  }

  ycur[b * T_ + tid] = y0;          ycur[b * T_ + tid + 1024] = y1;
  prev[b * T_ + tid] += y0;         prev[b * T_ + tid + 1024] += y1;
  size_t ob = ((size_t)b * LST + stage) * T_;
  out[ob + tid] = y0;               out[ob + tid + 1024] = y1;
}

// ------------- compactness: gram via WMMA, dist-weighted sums -------------
// Grid: B * (T/64) blocks of 128 (4 waves). Wave w handles one 16-row i-tile;
// its 16x128 A slab stays resident in 32 v2f fragments across the j-loop.
// j-tiles of u are staged once per block into LDS (async, double-buffered)
// and shared by all 4 waves.
__global__ void __launch_bounds__(128) k_compact(
    const float* __restrict__ u, const float* __restrict__ sq,
    const float* __restrict__ ycur, float* __restrict__ stats) {
  __shared__ float tile[2][16 * LDSROW];

  int tid = threadIdx.x;
  int wave = tid >> 5, lane = tid & 31;
  int b = blockIdx.x / (T_ / 64);
  int bt = blockIdx.x % (T_ / 64);
  int i0 = (bt * 4 + wave) * 16;
  int n16 = lane & 15, khalf = lane >> 4;
  const float* ub = u + (size_t)b * T_ * DU;

  // Stage one 16x128 j-tile into LDS: 512 x 16B transfers, 4 per thread.
  auto issue_tile = [&](int buf, int jt) {
    const float* gsrc = ub + (size_t)jt * 16 * DU;
#pragma unroll
    for (int c = 0; c < 4; ++c) {
      int i = c * 128 + tid;          // 0..511
      int row = i >> 5, seg = i & 31; // 16 rows x 32 16B-segments
      const float* g = gsrc + row * DU + seg * 4;
      float* l = &tile[buf][row * LDSROW + seg * 4];
#if HAVE_ASYNC_LDS
      __builtin_amdgcn_global_load_async_to_lds_b128(
          (AS1 v4i*)(uintptr_t)g,
          (AS3 v4i*)(unsigned)(uintptr_t)l,  // low 32b of generic = LDS offset
          0, 0);
#else
      *(v4f*)l = *(const v4f*)g;
#endif
    }
  };

  v2f afrag[32];
  {
    const float* ar = ub + (size_t)(i0 + n16) * DU + khalf * 2;
#pragma unroll
    for (int kk = 0; kk < 32; ++kk) afrag[kk] = *(const v2f*)(ar + kk * 4);
  }
  float sqm[8];
#pragma unroll
  for (int v = 0; v < 8; ++v) sqm[v] = sq[b * T_ + i0 + 8 * khalf + v];
  float wacc[8];
#pragma unroll
  for (int v = 0; v < 8; ++v) wacc[v] = 0.f;

  issue_tile(0, 0);
  for (int jt = 0; jt < T_ / 16; ++jt) {
    int cur = jt & 1;
    if (jt + 1 < T_ / 16) {
      issue_tile(cur ^ 1, jt + 1);     // prefetch next tile into other buffer
#if HAVE_ASYNC_LDS
      WAIT_ASYNC(4);                   // this tile's 4 copies retired (in-order)
#endif
    } else {
#if HAVE_ASYNC_LDS
      WAIT_ASYNC(0);
#endif
    }
    __syncthreads();

    int j0 = jt * 16;
    const float* brow = &tile[cur][n16 * LDSROW + khalf * 2];
    v8f acc = {};
#pragma unroll
    for (int kk = 0; kk < 32; ++kk) {
      v2f bfrag = *(const v2f*)(brow + kk * 4);
      acc = __builtin_amdgcn_wmma_f32_16x16x4_f32(false, afrag[kk], false, bfrag,
                                                  (short)0, acc, false, false);
    }
    float yn  = ycur[b * T_ + j0 + n16];
    float sqn = sq[b * T_ + j0 + n16];
#pragma unroll
    for (int v = 0; v < 8; ++v) {
      float d2 = fmaxf(sqm[v] + sqn - 2.f * acc[v], 0.f);
      wacc[v] += sqrtf(d2 + EPSV) * yn;
    }
    __syncthreads();   // all waves done reading tile[cur] before it is refilled
  }

  // sum over the 16 lanes of each half-wave (same M rows, different N columns)
#pragma unroll
  for (int v = 0; v < 8; ++v) {
    float w = wacc[v];
    w += __shfl_xor(w, 1);
    w += __shfl_xor(w, 2);
    w += __shfl_xor(w, 4);
    w += __shfl_xor(w, 8);
    wacc[v] = w;
  }
  if (n16 == 0) {
    float contrib = 0.f;
#pragma unroll
    for (int v = 0; v < 8; ++v)
      contrib += ycur[b * T_ + i0 + 8 * khalf + v] * wacc[v];
    atomicAdd(&stats[b * 8 + 4], contrib);
  }
}

// ---------------- z = (sum_t y_t u_t) @ W_v^T ----------------
__global__ void __launch_bounds__(128) k_z(
    const float* __restrict__ u, const float* __restrict__ ycur,
    const float* __restrict__ Wv, float* __restrict__ zbuf) {
  __shared__ float sv[DU];
  int b = blockIdx.x, tid = threadIdx.x;
  const float* ub = u + (size_t)b * T_ * DU + tid;
  const float* yb = ycur + b * T_;
  float s = 0.f;
  for (int t = 0; t < T_; ++t) s += yb[t] * ub[(size_t)t * DU];
  sv[tid] = s;
  __syncthreads();
  float zz = 0.f;
  for (int d = 0; d < DU; ++d) zz += sv[d] * Wv[tid * DU + d];
  zbuf[b * DU + tid] = zz;
}

// ---------------- GRU cell update ----------------
__global__ void __launch_bounds__(128) k_gru(
    const float* __restrict__ zbuf, const float* __restrict__ stats,
    const float* __restrict__ feedback,
    const float* __restrict__ Wih, const float* __restrict__ Whh,
    const float* __restrict__ bih, const float* __restrict__ bhh,
    float* __restrict__ mem, float* __restrict__ out_mem, int last) {
  __shared__ float gin[GRU_IN];
  __shared__ float hs[DM];
  int b = blockIdx.x, j = threadIdx.x;
  gin[j] = zbuf[b * DU + j];
  hs[j] = mem[b * DM + j];
  if (j == 0) {
    float total = stats[b * 8 + 0];
    gin[DU + 0] = stats[b * 8 + 1];
    gin[DU + 1] = stats[b * 8 + 2];
    gin[DU + 2] = stats[b * 8 + 3];
    gin[DU + 3] = stats[b * 8 + 4] / fmaxf(total * total, EPSV);
    gin[DU + 4] = feedback[b];
  }
  __syncthreads();
  float ir = bih[j], iz = bih[j + DM], inn = bih[j + 2 * DM];
  for (int k = 0; k < GRU_IN; ++k) {
    float g = gin[k];
    ir  += g * Wih[(size_t)j * GRU_IN + k];
    iz  += g * Wih[(size_t)(j + DM) * GRU_IN + k];
    inn += g * Wih[(size_t)(j + 2 * DM) * GRU_IN + k];
  }
  float hr = bhh[j], hz = bhh[j + DM], hn = bhh[j + 2 * DM];
  for (int k = 0; k < DM; ++k) {
    float h = hs[k];
    hr += h * Whh[(size_t)j * DM + k];
    hz += h * Whh[(size_t)(j + DM) * DM + k];
    hn += h * Whh[(size_t)(j + 2 * DM) * DM + k];
  }
  float rg = 1.f / (1.f + expf(-(ir + hr)));
  float zg = 1.f / (1.f + expf(-(iz + hz)));
  float ng = tanhf(inn + rg * hn);
  float hnew = (1.f - zg) * ng + zg * hs[j];
  mem[b * DM + j] = hnew;
  if (last) out_mem[b * DM + j] = hnew;
}

// ---------------- launch ----------------
extern "C" void kernel_launch(void* const* d_in, const int* in_sizes, int n_in,
                              void* d_out, int out_size, void* d_ws, size_t ws_size,
                              hipStream_t stream) {
  (void)in_sizes; (void)n_in; (void)out_size; (void)ws_size;
  const float* x        = (const float*)d_in[0];
  const float* feedback = (const float*)d_in[1];
  const float* W_u      = (const float*)d_in[2];
  const float* b_u      = (const float*)d_in[3];
  const float* W_q      = (const float*)d_in[4];
  const float* W_k      = (const float*)d_in[5];
  const float* W_v      = (const float*)d_in[6];
  const float* b_pos    = (const float*)d_in[7];
  const float* log_temp = (const float*)d_in[8];
  const float* Wih      = (const float*)d_in[9];
  const float* Whh      = (const float*)d_in[10];
  const float* bih      = (const float*)d_in[11];
  const float* bhh      = (const float*)d_in[12];
  float* out = (float*)d_out;
  float* ws  = (float*)d_ws;

  float* u     = ws;                        // B*T*DU
  float* sq    = u    + (size_t)B_ * T_ * DU;
  float* prev  = sq   + B_ * T_;
  float* ycur  = prev + B_ * T_;
  float* rbuf  = ycur + B_ * T_;            // B*DU
  float* zbuf  = rbuf + B_ * DU;            // B*DU
  float* stats = zbuf + B_ * DU;            // B*8
  float* mem   = stats + B_ * 8;            // B*DM
  float* out_mem = out + (size_t)B_ * LST * T_;

  k_init<<<(B_ * T_ + 255) / 256, 256, 0, stream>>>(prev, mem);
  k_proj_u<<<(B_ * T_ / 16) * (DU / 16) / 4, 128, 0, stream>>>(x, W_u, b_u, u);
  k_sq<<<(B_ * T_ + 255) / 256, 256, 0, stream>>>(u, sq);

  for (int stage = 0; stage < LST; ++stage) {
    k_qr<<<B_, 128, 0, stream>>>(mem, W_q, W_k, rbuf);
    k_select<<<B_, 1024, 0, stream>>>(u, rbuf, b_pos, log_temp, prev, ycur,
                                      stats, out, stage);
    k_compact<<<B_ * (T_ / 64), 128, 0, stream>>>(u, sq, ycur, stats);
    k_z<<<B_, 128, 0, stream>>>(u, ycur, W_v, zbuf);
    k_gru<<<B_, 128, 0, stream>>>(zbuf, stats, feedback, Wih, Whh, bih, bhh,
                                  mem, out_mem, stage == LST - 1);
  }
}